// MoEGate_66099546685735
// MI455X (gfx1250) — compile-verified
//
#include <hip/hip_runtime.h>

#define DIM       2048
#define NEXP      64
#define TOPK      8
#define TOK_WAVE  32
#define WAVES_BLK 4
#define TOK_BLK   (TOK_WAVE * WAVES_BLK)   // 128 tokens / block
#define N_TOKENS  32768                    // 4 * 8192
#define N_BLOCKS  (N_TOKENS / TOK_BLK)     // 256

typedef float v2f __attribute__((ext_vector_type(2)));
typedef float v8f __attribute__((ext_vector_type(8)));

// ---------------------------------------------------------------------------
// Kernel 1: fused gate GEMM (fp32 WMMA) + softmax + top-8 + histogram
// ---------------------------------------------------------------------------
__global__ __launch_bounds__(TOK_BLK) void moe_gate_main(
    const float* __restrict__ x,          // [T, DIM]
    const float* __restrict__ gw,         // [NEXP, DIM]
    const float* __restrict__ bias,       // [NEXP]
    float* __restrict__ out_w,            // [T, TOPK]
    float* __restrict__ out_idx,          // [T, TOPK] (indices as float)
    unsigned int* __restrict__ hist)      // [NEXP] global histogram
{
    __shared__ float        s_scores[TOK_BLK * NEXP];   // 32 KB
    __shared__ unsigned int s_hist[NEXP];

    const int tid  = threadIdx.x;
    const int lane = tid & 31;
    const int wave = tid >> 5;
    const int l16  = lane & 15;     // row / column within 16-wide tile
    const int half = lane >> 4;     // selects K pair per ISA A/B layout

    if (tid < NEXP) s_hist[tid] = 0u;

    const size_t tokBase = (size_t)blockIdx.x * TOK_BLK + (size_t)wave * TOK_WAVE;

    // A rows: token (M) = l16 within each 16-token tile
    const float* xr0 = x + (tokBase + l16) * (size_t)DIM;        // M-tile 0
    const float* xr1 = x + (tokBase + 16 + l16) * (size_t)DIM;   // M-tile 1
    // B columns: expert (N) = l16 within each 16-expert tile
    const float* gr  = gw + (size_t)l16 * DIM;

    v8f c[2][4];
    #pragma unroll
    for (int m = 0; m < 2; ++m)
        #pragma unroll
        for (int t = 0; t < 4; ++t)
            c[m][t] = v8f{0.f, 0.f, 0.f, 0.f, 0.f, 0.f, 0.f, 0.f};

    // K loop: V_WMMA_F32_16X16X4_F32, A layout => lanes 0-15 hold K=k..k+1,
    // lanes 16-31 hold K=k+2..k+3 -> one aligned b64 load per lane per tile.
    for (int k = 0; k < DIM; k += 4) {
        const int kk = k + 2 * half;
        const v2f a0 = *(const v2f*)(xr0 + kk);
        const v2f a1 = *(const v2f*)(xr1 + kk);
        v2f b[4];
        #pragma unroll
        for (int t = 0; t < 4; ++t)
            b[t] = *(const v2f*)(gr + (size_t)t * 16 * DIM + kk);
        #pragma unroll
        for (int t = 0; t < 4; ++t) {
            c[0][t] = __builtin_amdgcn_wmma_f32_16x16x4_f32(
                false, a0, false, b[t], (short)0, c[0][t], false, false);
            c[1][t] = __builtin_amdgcn_wmma_f32_16x16x4_f32(
                false, a1, false, b[t], (short)0, c[1][t], false, false);
        }
    }

    // Spill accumulators to LDS. C/D layout: VGPR j -> row M=j (lanes 0-15)
    // or M=j+8 (lanes 16-31), column N = l16.
    float* my = s_scores + wave * TOK_WAVE * NEXP;
    #pragma unroll
    for (int m = 0; m < 2; ++m)
        #pragma unroll
        for (int t = 0; t < 4; ++t)
            #pragma unroll
            for (int j = 0; j < 8; ++j)
                my[(m * 16 + j + 8 * half) * NEXP + (16 * t + l16)] = c[m][t][j];

    __syncthreads();

    // One thread per token: bias add -> softmax -> top-8 -> normalize.
    {
        const size_t gTok = (size_t)blockIdx.x * TOK_BLK + tid;
        float* sc = s_scores + tid * NEXP;

        float mx = -3.402823466e38f;
        #pragma unroll 8
        for (int e = 0; e < NEXP; ++e) {
            const float v = sc[e] + bias[e];
            sc[e] = v;
            mx = fmaxf(mx, v);
        }
        float sum = 0.f;
        #pragma unroll 8
        for (int e = 0; e < NEXP; ++e) {
            const float p = __expf(sc[e] - mx);
            sc[e] = p;
            sum += p;
        }
        const float inv = 1.0f / sum;

        float tw[TOPK];
        int   ti[TOPK];
        float tsum = 0.f;
        #pragma unroll
        for (int s = 0; s < TOPK; ++s) {
            float best = -1.f;
            int   bi   = 0;
            for (int e = 0; e < NEXP; ++e) {
                const float p = sc[e];
                if (p > best) { best = p; bi = e; }   // lowest index wins ties
            }
            sc[bi] = -1.f;
            const float p = best * inv;
            tw[s] = p;
            ti[s] = bi;
            tsum += p;
        }
        const float norm = 1.0f / (tsum + 1e-8f);
        #pragma unroll
        for (int s = 0; s < TOPK; ++s) {
            out_w[gTok * TOPK + s]   = tw[s] * norm;
            out_idx[gTok * TOPK + s] = (float)ti[s];
            atomicAdd(&s_hist[ti[s]], 1u);           // LDS histogram
        }
    }

    __syncthreads();
    if (tid < NEXP) atomicAdd(&hist[tid], s_hist[tid]);  // 1 global atomic/expert/block
}

// ---------------------------------------------------------------------------
// Kernel 2: finalize adaptive bias + usage EMA
// ---------------------------------------------------------------------------
__global__ void moe_gate_finalize(
    const unsigned int* __restrict__ hist,
    const float* __restrict__ bias,
    const float* __restrict__ old_usage,
    float* __restrict__ out_bias,
    float* __restrict__ out_usage)
{
    const int e = threadIdx.x;
    if (e < NEXP) {
        const float usage = (float)hist[e] * (1.0f / (float)(N_TOKENS * TOPK));
        out_bias[e]  = bias[e] - 0.01f * (usage - 1.0f / (float)NEXP);
        out_usage[e] = 0.9f * old_usage[e] + 0.1f * usage;
    }
}

// ---------------------------------------------------------------------------
extern "C" void kernel_launch(void* const* d_in, const int* in_sizes, int n_in,
                              void* d_out, int out_size, void* d_ws, size_t ws_size,
                              hipStream_t stream) {
    const float* x     = (const float*)d_in[0];   // [4,8192,2048]
    const float* gw    = (const float*)d_in[1];   // [64,2048]
    const float* bias  = (const float*)d_in[2];   // [64]
    const float* usage = (const float*)d_in[3];   // [64]

    float* out        = (float*)d_out;
    float* out_w      = out;                              // T*8
    float* out_idx    = out + (size_t)N_TOKENS * TOPK;    // T*8
    float* out_bias   = out + (size_t)2 * N_TOKENS * TOPK;
    float* out_usage  = out_bias + NEXP;

    unsigned int* hist = (unsigned int*)d_ws;

    hipMemsetAsync(hist, 0, NEXP * sizeof(unsigned int), stream);
    moe_gate_main<<<N_BLOCKS, TOK_BLK, 0, stream>>>(x, gw, bias, out_w, out_idx, hist);
    moe_gate_finalize<<<1, NEXP, 0, stream>>>(hist, bias, usage, out_bias, out_usage);
}